// ModelNet10Prototypes_25074019074118
// MI455X (gfx1250) — compile-verified
//
#include <hip/hip_runtime.h>
#include <math.h>

#define B_ROWS 131072
#define FDIM   256
#define NCAT   10
#define TEMP_INV (1.0f/0.07f)
#define PST    260            // padded row stride in LDS (bank-conflict-free)
#define TILES  (B_ROWS/16)

typedef __attribute__((ext_vector_type(2))) float v2f;
typedef __attribute__((ext_vector_type(4))) float v4f;
typedef __attribute__((ext_vector_type(8))) float v8f;

#define VZERO8 {0.f,0.f,0.f,0.f,0.f,0.f,0.f,0.f}

__device__ __forceinline__ float red_add16(float v) {
  v += __shfl_xor(v, 1, 32);
  v += __shfl_xor(v, 2, 32);
  v += __shfl_xor(v, 4, 32);
  v += __shfl_xor(v, 8, 32);
  return v;
}

// ---------------- kernel 0: zero workspace accumulators ----------------
__global__ void k0_zero(float* ws, int n) {
  for (int i = threadIdx.x + blockIdx.x * blockDim.x; i < n; i += blockDim.x * gridDim.x)
    ws[i] = 0.f;
}

// ---------------- kernel 1: per-category sums of normalized rows -------
#define ROWCASE(i) case i: a##i += f; c##i += 1; break;
#define MERGE(i) do {                                                        \
    _Pragma("unroll")                                                        \
    for (int j = 0; j < 8; ++j)                                              \
      atomicAdd(&lsum[(i)*FDIM + lane*8 + j], a##i[j]);                      \
    if (lane == 0) atomicAdd(&lcnt[(i)], (float)c##i);                       \
  } while (0)

__global__ __launch_bounds__(256) void k1_segsum(const float* __restrict__ x,
                                                 const int* __restrict__ cats,
                                                 float* __restrict__ gsum,
                                                 float* __restrict__ gcnt) {
  __shared__ float lsum[NCAT * FDIM];
  __shared__ float lcnt[NCAT];
  for (int i = threadIdx.x; i < NCAT * FDIM; i += blockDim.x) lsum[i] = 0.f;
  if (threadIdx.x < NCAT) lcnt[threadIdx.x] = 0.f;
  __syncthreads();

  const int lane = threadIdx.x & 31;
  const int wib  = threadIdx.x >> 5;
  const int gw   = blockIdx.x * 8 + wib;
  const int nw   = gridDim.x * 8;

  v8f a0=VZERO8,a1=VZERO8,a2=VZERO8,a3=VZERO8,a4=VZERO8;
  v8f a5=VZERO8,a6=VZERO8,a7=VZERO8,a8=VZERO8,a9=VZERO8;
  int c0=0,c1=0,c2=0,c3=0,c4=0,c5=0,c6=0,c7=0,c8=0,c9=0;

  for (int row = gw; row < B_ROWS; row += nw) {
    const float* xr = x + (size_t)row * FDIM + lane * 8;
    v4f u1 = *(const v4f*)xr;
    v4f u2 = *(const v4f*)(xr + 4);
    float ss = u1.x*u1.x + u1.y*u1.y + u1.z*u1.z + u1.w*u1.w
             + u2.x*u2.x + u2.y*u2.y + u2.z*u2.z + u2.w*u2.w;
    ss = red_add16(ss);
    ss += __shfl_xor(ss, 16, 32);
    float inv = 1.f / fmaxf(sqrtf(ss), 1e-12f);
    v8f f;
    f[0]=u1.x*inv; f[1]=u1.y*inv; f[2]=u1.z*inv; f[3]=u1.w*inv;
    f[4]=u2.x*inv; f[5]=u2.y*inv; f[6]=u2.z*inv; f[7]=u2.w*inv;
    int c = cats[row];                        // wave-uniform
    switch (c) {
      ROWCASE(0) ROWCASE(1) ROWCASE(2) ROWCASE(3) ROWCASE(4)
      ROWCASE(5) ROWCASE(6) ROWCASE(7) ROWCASE(8) ROWCASE(9)
      default: break;
    }
  }
  MERGE(0); MERGE(1); MERGE(2); MERGE(3); MERGE(4);
  MERGE(5); MERGE(6); MERGE(7); MERGE(8); MERGE(9);
  __syncthreads();
  for (int i = threadIdx.x; i < NCAT * FDIM; i += blockDim.x)
    atomicAdd(&gsum[i], lsum[i]);
  if (threadIdx.x < NCAT) atomicAdd(&gcnt[threadIdx.x], lcnt[threadIdx.x]);
}

// ---------------- kernel 2: EMA prototype update -----------------------
__global__ __launch_bounds__(512) void k2_proto(const float* __restrict__ gsum,
                                                const float* __restrict__ gcnt,
                                                const float* __restrict__ proto,
                                                float* __restrict__ pw,
                                                float* __restrict__ pn2) {
  const int w    = threadIdx.x >> 5;
  const int lane = threadIdx.x & 31;
  if (w >= NCAT) return;
  float cnt  = gcnt[w];
  float invc = 1.f / fmaxf(cnt, 1.f);
  const float* s = gsum  + w * FDIM + lane * 8;
  const float* p = proto + w * FDIM + lane * 8;
  float e[8], pp[8];
  float ss = 0.f;
  #pragma unroll
  for (int j = 0; j < 8; ++j) {
    pp[j] = p[j];
    float ev = 0.9f * pp[j] + 0.1f * (s[j] * invc);
    e[j] = ev;
    ss += ev * ev;
  }
  ss = red_add16(ss);
  ss += __shfl_xor(ss, 16, 32);
  float inv = 1.f / fmaxf(sqrtf(ss), 1e-12f);
  bool has = cnt > 0.f;
  float os = 0.f;
  #pragma unroll
  for (int j = 0; j < 8; ++j) {
    float ov = has ? e[j] * inv : pp[j];
    pw[w * FDIM + lane * 8 + j] = ov;
    os += ov * ov;
  }
  os = red_add16(os);
  os += __shfl_xor(os, 16, 32);
  if (lane == 0) pn2[w] = os;
}

// ---------------- kernel 3: WMMA sim + log-softmax + aligned -----------
// K-permuted chunk pairs: one b128 per lane feeds TWO wmma ops.
//   wmma #1 of a pair uses physical k {kk,kk+1} (h=0) / {kk+4,kk+5} (h=1)
//   wmma #2 of a pair uses physical k {kk+2,kk+3} (h=0) / {kk+6,kk+7} (h=1)
// Same permutation applied to A (features) and B (p_used^T) => exact dot.
__global__ __launch_bounds__(256) void k3_main(const float* __restrict__ x,
                                               const int* __restrict__ cats,
                                               const float* __restrict__ pw,
                                               const float* __restrict__ pn2,
                                               float* __restrict__ lossacc,
                                               float* __restrict__ out1) {
  __shared__ float pl[16 * PST];   // p_used padded to 16 rows, stride 260
  __shared__ float prm[8][48];     // per-wave row params: invN[16], invA[16], cat[16]
  __shared__ float pn2s[NCAT];

  for (int i = threadIdx.x; i < 16 * PST; i += blockDim.x) {
    int rr = i / PST, k = i - rr * PST;
    pl[i] = (rr < NCAT && k < FDIM) ? pw[rr * FDIM + k] : 0.f;
  }
  if (threadIdx.x < NCAT) pn2s[threadIdx.x] = pn2[threadIdx.x];
  __syncthreads();

  const int lane = threadIdx.x & 31;
  const int wib  = threadIdx.x >> 5;
  const int h    = lane >> 4;      // half-wave
  const int r    = lane & 15;      // row (A) / column (B)
  const int gw   = blockIdx.x * 8 + wib;
  const int nw   = gridDim.x * 8;
  float* myp = prm[wib];
  float lsum = 0.f;

  for (int tile = gw; tile < TILES; tile += nw) {
    const int rowbase = tile * 16;
    const float* xrow = x  + (size_t)(rowbase + r) * FDIM + 4 * h;
    const float* brow = pl + r * PST + 4 * h;
    v8f acc = VZERO8;
    float ssq = 0.f;
    #pragma unroll 4
    for (int kk = 0; kk < FDIM; kk += 8) {
      v4f a = *(const v4f*)(xrow + kk);   // b128: feeds 2 wmma chunks
      v4f b = *(const v4f*)(brow + kk);
      ssq = fmaf(a.x, a.x, ssq);
      ssq = fmaf(a.y, a.y, ssq);
      ssq = fmaf(a.z, a.z, ssq);
      ssq = fmaf(a.w, a.w, ssq);
      v2f a0 = __builtin_shufflevector(a, a, 0, 1);
      v2f a1 = __builtin_shufflevector(a, a, 2, 3);
      v2f b0 = __builtin_shufflevector(b, b, 0, 1);
      v2f b1 = __builtin_shufflevector(b, b, 2, 3);
      acc = __builtin_amdgcn_wmma_f32_16x16x4_f32(
          false, a0, false, b0, (short)0, acc, false, false);
      acc = __builtin_amdgcn_wmma_f32_16x16x4_f32(
          false, a1, false, b1, (short)0, acc, false, false);
    }
    // full row ||x||^2: lanes r and r+16 cover complementary K subsets
    ssq += __shfl_xor(ssq, 16, 32);
    float invn = 1.f / fmaxf(sqrtf(ssq), 1e-12f);
    if (h == 0) myp[r] = invn;
    int cid = cats[rowbase + r];

    #pragma unroll
    for (int j = 0; j < 8; ++j) {
      const int M = j + 8 * h;                 // output row of this acc VGPR
      float invM = __shfl(invn, M, 32);
      int   cM   = __shfl(cid,  M, 32);
      float dotn = acc[j] * invM;              // f . p_c  (normalized)
      float sim  = dotn * TEMP_INV;
      float vmax = (r < NCAT) ? sim : -3.0e38f;
      vmax = fmaxf(vmax, __shfl_xor(vmax, 1, 32));
      vmax = fmaxf(vmax, __shfl_xor(vmax, 2, 32));
      vmax = fmaxf(vmax, __shfl_xor(vmax, 4, 32));
      vmax = fmaxf(vmax, __shfl_xor(vmax, 8, 32));
      float ex = (r < NCAT) ? expf(sim - vmax) : 0.f;
      ex = red_add16(ex);
      float dc = (r == cM) ? dotn : 0.f;       // dot at target category
      dc = red_add16(dc);
      float lse = vmax + logf(ex);
      if (r == 0) {                            // lanes 0 and 16: rows 0-7 / 8-15
        lsum += lse - dc * TEMP_INV;           // -logp[i, cat]
        // ||0.7 f + 0.3 p_c||^2 = 0.49 + 0.09||p_c||^2 + 0.42 (f.p_c)
        float a2 = 0.49f + 0.09f * pn2s[cM] + 0.42f * dc;
        myp[16 + M] = 1.f / fmaxf(sqrtf(a2), 1e-12f);
        myp[32 + M] = (float)cM;
      }
    }

    // write phase: aligned = (0.7*x*invN + 0.3*p_cat) * invA
    // x re-read hits L2 (NT stores below keep features resident).
    for (int rr = 0; rr < 16; ++rr) {
      float invN = myp[rr];
      float invA = myp[16 + rr];
      int   c    = (int)myp[32 + rr];
      const float* xg = x    + (size_t)(rowbase + rr) * FDIM;
      const float* pr = pl   + c * PST;
      float*       og = out1 + (size_t)(rowbase + rr) * FDIM;
      #pragma unroll
      for (int jj = 0; jj < 8; ++jj) {
        int k = lane + 32 * jj;
        float val = (0.7f * xg[k] * invN + 0.3f * pr[k]) * invA;
        __builtin_nontemporal_store(val, &og[k]);   // NT: don't pollute L2
      }
    }
  }
  lsum += __shfl_xor(lsum, 16, 32);
  if (lane == 0) atomicAdd(lossacc, lsum);
}

// ---------------- kernel 4: finalize loss ------------------------------
__global__ void k4_final(const float* __restrict__ lossacc, float* __restrict__ out) {
  if (threadIdx.x == 0 && blockIdx.x == 0)
    out[0] = lossacc[0] / (float)B_ROWS;
}

extern "C" void kernel_launch(void* const* d_in, const int* in_sizes, int n_in,
                              void* d_out, int out_size, void* d_ws, size_t ws_size,
                              hipStream_t stream) {
  (void)in_sizes; (void)n_in; (void)out_size; (void)ws_size;
  const float* x     = (const float*)d_in[0];
  const int*   cats  = (const int*)d_in[1];
  const float* proto = (const float*)d_in[2];
  float* out = (float*)d_out;
  float* ws  = (float*)d_ws;

  float* gsum    = ws;           // 2560 floats
  float* gcnt    = ws + 2560;    // 10
  float* lossacc = ws + 2570;    // 1
  float* pw      = ws + 2576;    // 2560 (p_used)
  float* pn2     = ws + 5136;    // 10   (||p_used_c||^2)

  k0_zero  <<<8,    256, 0, stream>>>(ws, 2571);
  k1_segsum<<<256,  256, 0, stream>>>(x, cats, gsum, gcnt);
  k2_proto <<<1,    512, 0, stream>>>(gsum, gcnt, proto, pw, pn2);
  k3_main  <<<1024, 256, 0, stream>>>(x, cats, pw, pn2, lossacc, out + 1);
  k4_final <<<1,    32,  0, stream>>>(lossacc, out);
}